// BufferRetrievalHungarianMatcher_12352325943617
// MI455X (gfx1250) — compile-verified
//
#include <hip/hip_runtime.h>

typedef float v2f __attribute__((ext_vector_type(2)));
typedef float v8f __attribute__((ext_vector_type(8)));

#define BATCH 16
#define NDIM  256    // N = O = 256 (square assignment)
#define MDIM  2048   // feature dim
#define NP    288    // padded 257 (sentinel col 0 + 256 cols), 32 lanes * 9
#define KOWN  9      // columns owned per lane: j = lane + 32*k

// ---------------------------------------------------------------------------
// Kernel 1: cost[b,i,j] = -dot(outputs[b,i,:], targets[b,j,:])  via f32 WMMA.
// One wave computes a 16x64 tile: 1 A-fragment load + 4 B-fragment loads +
// 4 v_wmma_f32_16x16x4_f32 per K-step.  f32 A(16x4)/B(4x16) fragment layout:
//   lane l: row/col = l&15, VGPR0 = K=(l>>4)*2, VGPR1 = K=(l>>4)*2+1
// which for row-major [256][2048] storage is one contiguous b64 load per lane.
// ---------------------------------------------------------------------------
__global__ __launch_bounds__(256) void cost_gemm_wmma(
    const float* __restrict__ outs, const float* __restrict__ tgts,
    float* __restrict__ cost)
{
  const int wave = (int)((blockIdx.x * blockDim.x + threadIdx.x) >> 5);
  const int lane = (int)(threadIdx.x & 31u);
  const int lo   = lane & 15;
  const int hi   = lane >> 4;

  const int jt = wave & 3;          // which 64-col group
  const int it = (wave >> 2) & 15;  // which 16-row tile
  const int b  = wave >> 6;         // batch

  const float* Ap = outs + ((size_t)(b * NDIM + it * 16 + lo)) * MDIM + hi * 2;
  const float* Bp = tgts + ((size_t)(b * NDIM + jt * 64 + lo)) * MDIM + hi * 2;

  v8f acc0 = {}, acc1 = {}, acc2 = {}, acc3 = {};
#pragma unroll 4
  for (int kk = 0; kk < MDIM; kk += 4) {
    v2f a  = *(const v2f*)(Ap + kk);
    v2f b0 = *(const v2f*)(Bp + kk);
    v2f b1 = *(const v2f*)(Bp + 16 * MDIM + kk);
    v2f b2 = *(const v2f*)(Bp + 32 * MDIM + kk);
    v2f b3 = *(const v2f*)(Bp + 48 * MDIM + kk);
    acc0 = __builtin_amdgcn_wmma_f32_16x16x4_f32(false, a, false, b0, (short)0, acc0, false, false);
    acc1 = __builtin_amdgcn_wmma_f32_16x16x4_f32(false, a, false, b1, (short)0, acc1, false, false);
    acc2 = __builtin_amdgcn_wmma_f32_16x16x4_f32(false, a, false, b2, (short)0, acc2, false, false);
    acc3 = __builtin_amdgcn_wmma_f32_16x16x4_f32(false, a, false, b3, (short)0, acc3, false, false);
  }

  // C/D layout: element r -> row it*16 + r + 8*hi, col jt*64 + q*16 + lo
  float* Cb = cost + (size_t)b * NDIM * NDIM;
  v8f accs[4] = {acc0, acc1, acc2, acc3};
#pragma unroll
  for (int q = 0; q < 4; ++q) {
    const int j = jt * 64 + q * 16 + lo;
#pragma unroll
    for (int r = 0; r < 8; ++r) {
      const int i = it * 16 + r + 8 * hi;
      Cb[i * NDIM + j] = -accs[q][r];
    }
  }
}

// ---------------------------------------------------------------------------
// Kernel 2: exact JV Hungarian, one single-wave workgroup per batch.
// Entire 256KB cost matrix lives in LDS (320KB/WGP on CDNA5).  Lane l owns
// columns j = l + 32k (k<9); cost-row reads are stride-1 across lanes.
// Wave-level argmin via __shfl_xor (wave32).  out is int32 [B,2,N].
// ---------------------------------------------------------------------------
__global__ __launch_bounds__(32) void hungarian_lds(
    const float* __restrict__ cost_g, int* __restrict__ out)
{
  extern __shared__ char smem[];
  float* C    = (float*)smem;                                  // [256*256]
  float* u    = (float*)(smem + NDIM * NDIM * sizeof(float));  // [NP]
  float* v    = u + NP;
  float* minv = v + NP;
  int*   way  = (int*)(minv + NP);
  int*   used = way + NP;
  int*   p    = used + NP;

  const int b    = (int)blockIdx.x;
  const int lane = (int)threadIdx.x;
  const float INF = __builtin_inff();

  // stage cost matrix into LDS (b128 loads/stores)
  {
    const float4* src = (const float4*)(cost_g + (size_t)b * NDIM * NDIM);
    float4* dst = (float4*)C;
    for (int idx = lane; idx < (NDIM * NDIM) / 4; idx += 32) dst[idx] = src[idx];
  }
#pragma unroll
  for (int k = 0; k < KOWN; ++k) {
    int j = lane + 32 * k;
    u[j] = 0.0f; v[j] = 0.0f; p[j] = 0;
  }
  __syncthreads();

  for (int i = 1; i <= NDIM; ++i) {
#pragma unroll
    for (int k = 0; k < KOWN; ++k) {
      int j = lane + 32 * k;
      minv[j] = INF; way[j] = 0; used[j] = 0;
    }
    if (lane == 0) p[0] = i;
    __syncthreads();

    int j0 = 0;
    for (;;) {
      const int i0 = p[j0];           // uniform broadcast read
      if (i0 == 0) break;             // found a free column -> augment
      if (lane == 0) used[j0] = 1;
      __syncthreads();

      const float ui0 = u[i0];
      const float* Crow = C + (i0 - 1) * NDIM;
      float best = INF; int bestj = 0;
#pragma unroll
      for (int k = 0; k < KOWN; ++k) {
        const int j = lane + 32 * k;
        if (j >= 1 && j <= NDIM && !used[j]) {
          float mv  = minv[j];
          float cur = Crow[j - 1] - ui0 - v[j];
          if (cur < mv) { mv = cur; minv[j] = cur; way[j] = j0; }
          if (mv < best) { best = mv; bestj = j; }
        }
      }
      // wave32 argmin (lowest index on ties, matching argmin-first semantics)
#pragma unroll
      for (int off = 16; off >= 1; off >>= 1) {
        float ov = __shfl_xor(best, off, 32);
        int   oj = __shfl_xor(bestj, off, 32);
        if (ov < best || (ov == best && oj < bestj)) { best = ov; bestj = oj; }
      }
      const float delta = best;
      const int j1 = bestj;
      __syncthreads();

      // dual updates; p[j] distinct over used j => race-free
#pragma unroll
      for (int k = 0; k < KOWN; ++k) {
        const int j = lane + 32 * k;
        if (j <= NDIM) {
          if (used[j]) { const int pj = p[j]; u[pj] = u[pj] + delta; v[j] -= delta; }
          else         { minv[j] -= delta; }
        }
      }
      __syncthreads();
      j0 = j1;
    }

    // augment along the alternating tree (serial, lane 0)
    if (lane == 0) {
      int jj = j0;
      while (jj != 0) { const int jprev = way[jj]; p[jj] = p[jprev]; jj = jprev; }
    }
    __syncthreads();
  }

  // invert matching: col j-1 assigned to row p[j]-1 ; output [2,N] per batch
  int* outb = out + b * 2 * NDIM;
#pragma unroll
  for (int k = 0; k < KOWN; ++k) {
    const int j = lane + 32 * k;
    if (j >= 1 && j <= NDIM) {
      const int row = p[j] - 1;
      outb[row] = row;              // row indices (arange)
      outb[NDIM + row] = j - 1;     // matched column
    }
  }
}

extern "C" void kernel_launch(void* const* d_in, const int* in_sizes, int n_in,
                              void* d_out, int out_size, void* d_ws, size_t ws_size,
                              hipStream_t stream) {
  const float* outputs = (const float*)d_in[0];   // [16,256,2048] f32
  const float* targets = (const float*)d_in[1];   // [16,256,2048] f32
  int*   out  = (int*)d_out;                      // [16,2,256] i32
  float* cost = (float*)d_ws;                     // [16,256,256] f32 scratch

  // GEMM: 16 batches * 16 i-tiles * 4 j-groups = 1024 waves = 128 blocks of 256
  cost_gemm_wmma<<<128, 256, 0, stream>>>(outputs, targets, cost);

  const size_t smem = (size_t)NDIM * NDIM * sizeof(float)
                    + 3 * NP * sizeof(float) + 3 * NP * sizeof(int); // ~269KB < 320KB
  (void)hipFuncSetAttribute(reinterpret_cast<const void*>(&hungarian_lds),
                            hipFuncAttributeMaxDynamicSharedMemorySize, (int)smem);
  hungarian_lds<<<BATCH, 32, smem, stream>>>(cost, out);
}